// Co_VGAE_25752623907299
// MI455X (gfx1250) — compile-verified
//
#include <hip/hip_runtime.h>
#include <hip/hip_bf16.h>

typedef float v2f __attribute__((ext_vector_type(2)));
typedef float v8f __attribute__((ext_vector_type(8)));

#define N_FEAT_IN 128
#define DLAT 64

__device__ __forceinline__ float sigf(float x) {
  return 1.0f / (1.0f + __expf(-x));
}

// ---------------------------------------------------------------------------
// Generic WMMA fp32 GEMM: C[M x Nc] = A[M x K] @ B[K x Nc] (row-major).
// One wave (32 threads) per 16x16 output tile, K-loop in steps of 4 using
// v_wmma_f32_16x16x4_f32.
// EPI: 0 = none, 1 = +bias, 2 = +bias then LeakyReLU(0.01)
// Fragment layouts per CDNA5 ISA 7.12.2:
//   A 16x4 : lanes 0-15 -> M=lane, K={0,1}; lanes 16-31 -> M=lane-16, K={2,3}
//   B 4x16 : lanes 0-15 -> N=lane, K={0,1}; lanes 16-31 -> N=lane-16, K={2,3}
//   C/D    : vgpr j -> (M = j + 8*half, N = lane&15)
// ---------------------------------------------------------------------------
template <int EPI>
__global__ void wmma_gemm16(const float* __restrict__ A,
                            const float* __restrict__ B,
                            const float* __restrict__ bias,
                            float* __restrict__ C,
                            int M, int K, int Nc) {
  const int lane = threadIdx.x & 31;
  const int hh   = lane >> 4;       // lane-half select
  const int l    = lane & 15;
  const int tn   = blockIdx.x * 16; // output col tile
  const int tm   = blockIdx.y * 16; // output row tile

  v8f acc = {};
  const int arow = tm + l;
  const int bcol = tn + l;
  for (int k = 0; k < K; k += 4) {
    v2f a, b;
    const int ka = k + hh * 2;
    a.x = A[(size_t)arow * K + ka + 0];
    a.y = A[(size_t)arow * K + ka + 1];
    b.x = B[(size_t)(ka + 0) * Nc + bcol];
    b.y = B[(size_t)(ka + 1) * Nc + bcol];
    acc = __builtin_amdgcn_wmma_f32_16x16x4_f32(false, a, false, b,
                                                (short)0, acc, false, false);
  }

  const int n = tn + l;
#pragma unroll
  for (int j = 0; j < 8; ++j) {
    const int m = tm + hh * 8 + j;
    float v = acc[j];
    if (EPI >= 1) v += bias[n];
    if (EPI == 2) v = (v > 0.0f) ? v : 0.01f * v;
    if (m < M && n < Nc) C[(size_t)m * Nc + n] = v;
  }
}

// ---------------------------------------------------------------------------
// A_pred = sigmoid(z @ z^T), z: [N x 64].  LDS-blocked fp32 WMMA.
// Block = 256 threads (8 waves). Output tile 128 rows x 64 cols per block.
// Each wave: 16 rows x 64 cols = 4 WMMA column tiles, K=64 in 16 steps.
// LDS row stride 68 floats: bank-conflict-free + 16B aligned float4 fills.
// ---------------------------------------------------------------------------
#define LDPAD 68
__global__ void __launch_bounds__(256)
zzt_sigmoid_kernel(const float* __restrict__ z, float* __restrict__ out, int N) {
  __shared__ float As[128 * LDPAD];
  __shared__ float Bs[64 * LDPAD];

  const int tid     = threadIdx.x;
  const int wave    = tid >> 5;
  const int lane    = tid & 31;
  const int hh      = lane >> 4;
  const int l       = lane & 15;
  const int rowBase = blockIdx.y * 128;
  const int colBase = blockIdx.x * 64;

  // Stage A rows (128 x 64) and B rows (= output columns, 64 x 64) into LDS.
#pragma unroll
  for (int i = 0; i < 8; ++i) {
    const int idx4 = tid + i * 256;        // 2048 float4 total
    const int r    = idx4 >> 4;            // 16 float4 per 64-float row
    const int c4   = idx4 & 15;
    const int gr   = min(rowBase + r, N - 1);
    *(float4*)&As[r * LDPAD + c4 * 4] =
        *(const float4*)&z[(size_t)gr * DLAT + c4 * 4];
  }
#pragma unroll
  for (int i = 0; i < 4; ++i) {
    const int idx4 = tid + i * 256;        // 1024 float4 total
    const int r    = idx4 >> 4;
    const int c4   = idx4 & 15;
    const int gr   = min(colBase + r, N - 1);
    *(float4*)&Bs[r * LDPAD + c4 * 4] =
        *(const float4*)&z[(size_t)gr * DLAT + c4 * 4];
  }
  __syncthreads();

  v8f acc[4] = {{}, {}, {}, {}};
  const int arow = wave * 16 + l;
#pragma unroll
  for (int k = 0; k < DLAT; k += 4) {
    const int ka = k + hh * 2;
    v2f a;
    a.x = As[arow * LDPAD + ka + 0];
    a.y = As[arow * LDPAD + ka + 1];
#pragma unroll
    for (int t = 0; t < 4; ++t) {
      v2f b;
      b.x = Bs[(t * 16 + l) * LDPAD + ka + 0];
      b.y = Bs[(t * 16 + l) * LDPAD + ka + 1];
      acc[t] = __builtin_amdgcn_wmma_f32_16x16x4_f32(false, a, false, b,
                                                     (short)0, acc[t],
                                                     false, false);
    }
  }

#pragma unroll
  for (int t = 0; t < 4; ++t) {
    const int col = colBase + t * 16 + l;
    if (col < N) {
#pragma unroll
      for (int j = 0; j < 8; ++j) {
        const int row = rowBase + wave * 16 + hh * 8 + j;
        if (row < N) out[(size_t)row * N + col] = sigf(acc[t][j]);
      }
    }
  }
}

// ---------------------------------------------------------------------------
// Graph / elementwise kernels
// ---------------------------------------------------------------------------
__global__ void k_deg_init(float* deg, int n) {
  int i = blockIdx.x * blockDim.x + threadIdx.x;
  if (i < n) deg[i] = 1.0f;  // self-loop
}

__global__ void k_deg_accum(const int* __restrict__ dst, float* deg, int e) {
  int i = blockIdx.x * blockDim.x + threadIdx.x;
  if (i < e) atomicAdd(&deg[dst[i]], 1.0f);
}

__global__ void k_dis(float* deg, int n) {  // in place: deg -> rsqrt(deg)
  int i = blockIdx.x * blockDim.x + threadIdx.x;
  if (i < n) deg[i] = rsqrtf(deg[i]);
}

// agg[i,f] = h[i,f] * dis[i]^2   (self-loop term)
__global__ void k_selfinit(const float* __restrict__ h,
                           const float* __restrict__ dis,
                           float* __restrict__ agg, int n) {
  int idx = blockIdx.x * blockDim.x + threadIdx.x;
  if (idx < n * DLAT) {
    int i = idx >> 6;
    float d = dis[i];
    agg[idx] = h[idx] * d * d;
  }
}

// agg[dst,f] += h[src,f] * dis[src]*dis[dst]   (64 threads per edge)
__global__ void k_scatter(const float* __restrict__ h,
                          const float* __restrict__ dis,
                          const int* __restrict__ src,
                          const int* __restrict__ dst,
                          float* __restrict__ agg, int e) {
  long long t = (long long)blockIdx.x * blockDim.x + threadIdx.x;
  int ei = (int)(t >> 6);
  int f  = (int)(t & 63);
  if (ei < e) {
    int s = src[ei], d = dst[ei];
    float w = dis[s] * dis[d];
    atomicAdd(&agg[(size_t)d * DLAT + f], h[(size_t)s * DLAT + f] * w);
  }
}

__global__ void k_zero(float* p, int n) {
  int i = blockIdx.x * blockDim.x + threadIdx.x;
  if (i < n) p[i] = 0.0f;
}

// In-place sigmoid(agg + bias) and accumulate per-feature sum/sumsq.
// blockDim = 64 (thread = feature), rows strided by grid.
__global__ void k_sig_stats(float* __restrict__ agg,
                            const float* __restrict__ bias,
                            float* __restrict__ sums, int n) {
  int f = threadIdx.x;
  float b = bias[f];
  float s = 0.0f, s2 = 0.0f;
  for (int r = blockIdx.x; r < n; r += gridDim.x) {
    float t = sigf(agg[(size_t)r * DLAT + f] + b);
    agg[(size_t)r * DLAT + f] = t;
    s += t;
    s2 += t * t;
  }
  atomicAdd(&sums[f], s);
  atomicAdd(&sums[DLAT + f], s2);
}

// out = (act - mu) * rsqrt(var + eps) * gamma + beta  (biased variance)
__global__ void k_bn_apply(const float* __restrict__ agg,
                           const float* __restrict__ sums,
                           const float* __restrict__ gamma,
                           const float* __restrict__ beta,
                           float* __restrict__ out, int n) {
  int idx = blockIdx.x * blockDim.x + threadIdx.x;
  if (idx < n * DLAT) {
    int f = idx & 63;
    float inv_n = 1.0f / (float)n;
    float mu  = sums[f] * inv_n;
    float var = sums[DLAT + f] * inv_n - mu * mu;
    out[idx] = (agg[idx] - mu) * rsqrtf(var + 1e-4f) * gamma[f] + beta[f];
  }
}

// z = noise * exp(logstd) + mean
__global__ void k_reparam(const float* __restrict__ noise,
                          const float* __restrict__ logstd,
                          const float* __restrict__ meanb,
                          float* __restrict__ z, int total) {
  int i = blockIdx.x * blockDim.x + threadIdx.x;
  if (i < total) z[i] = noise[i] * __expf(logstd[i]) + meanb[i];
}

// ---------------------------------------------------------------------------
// Host launcher
// ---------------------------------------------------------------------------
extern "C" void kernel_launch(void* const* d_in, const int* in_sizes, int n_in,
                              void* d_out, int out_size, void* d_ws, size_t ws_size,
                              hipStream_t stream) {
  (void)n_in; (void)out_size; (void)ws_size;

  const float* x     = (const float*)d_in[0];
  const int*   src   = (const int*)d_in[1];
  const int*   dst   = (const int*)d_in[2];
  /* d_in[3] = batch (unused) */
  const float* noise = (const float*)d_in[4];
  const float* W0 = (const float*)d_in[5];  const float* b0 = (const float*)d_in[6];
  const float* g0 = (const float*)d_in[7];  const float* be0 = (const float*)d_in[8];
  const float* Wm = (const float*)d_in[9];  const float* bm = (const float*)d_in[10];
  const float* gm = (const float*)d_in[11]; const float* bem = (const float*)d_in[12];
  const float* Ws = (const float*)d_in[13]; const float* bs = (const float*)d_in[14];
  const float* gs = (const float*)d_in[15]; const float* bes = (const float*)d_in[16];
  const float* Dw1 = (const float*)d_in[17]; const float* Db1 = (const float*)d_in[18];
  const float* Dw2 = (const float*)d_in[19]; const float* Db2 = (const float*)d_in[20];

  const int N = in_sizes[0] / N_FEAT_IN;
  const int E = in_sizes[1];

  float* ws = (float*)d_ws;
  size_t o = 0;
  float* dis    = ws + o; o += (size_t)N;
  float* htmp   = ws + o; o += (size_t)N * DLAT;
  float* agg    = ws + o; o += (size_t)N * DLAT;
  float* sums   = ws + o; o += 2 * DLAT;
  float* hidden = ws + o; o += (size_t)N * DLAT;
  float* meanb  = ws + o; o += (size_t)N * DLAT;
  float* logstd = ws + o; o += (size_t)N * DLAT;
  float* zb     = ws + o; o += (size_t)N * DLAT;
  float* h1     = ws + o; o += (size_t)N * DLAT;

  float* recon  = (float*)d_out;                       // [N x 128]
  float* A_pred = (float*)d_out + (size_t)N * N_FEAT_IN; // [N x N]

  const int ND = N * DLAT;

  // Degree / normalization (graph is shared by all three GCN layers)
  k_deg_init<<<(N + 255) / 256, 256, 0, stream>>>(dis, N);
  k_deg_accum<<<(E + 255) / 256, 256, 0, stream>>>(dst, dis, E);
  k_dis<<<(N + 255) / 256, 256, 0, stream>>>(dis, N);

  // Encoder layer: GCNConv -> sigmoid -> BN(train)
  auto enc_layer = [&](const float* in, int K, const float* W, const float* b,
                       const float* g, const float* be, float* outbuf) {
    dim3 gg((DLAT + 15) / 16, (N + 15) / 16);
    wmma_gemm16<0><<<gg, 32, 0, stream>>>(in, W, nullptr, htmp, N, K, DLAT);
    k_selfinit<<<(ND + 255) / 256, 256, 0, stream>>>(htmp, dis, agg, N);
    long long st = (long long)E * 64;
    k_scatter<<<(unsigned)((st + 255) / 256), 256, 0, stream>>>(htmp, dis, src, dst, agg, E);
    k_zero<<<1, 2 * DLAT, 0, stream>>>(sums, 2 * DLAT);
    k_sig_stats<<<128, DLAT, 0, stream>>>(agg, b, sums, N);
    k_bn_apply<<<(ND + 255) / 256, 256, 0, stream>>>(agg, sums, g, be, outbuf, N);
  };

  enc_layer(x,      N_FEAT_IN, W0, b0, g0, be0, hidden);
  enc_layer(hidden, DLAT,      Wm, bm, gm, bem, meanb);
  enc_layer(hidden, DLAT,      Ws, bs, gs, bes, logstd);

  // Reparameterization
  k_reparam<<<(ND + 255) / 256, 256, 0, stream>>>(noise, logstd, meanb, zb, ND);

  // Decoder MLP: h1 = leaky_relu(z @ Dw1 + Db1); recon = h1 @ Dw2 + Db2
  {
    dim3 g1((DLAT + 15) / 16, (N + 15) / 16);
    wmma_gemm16<2><<<g1, 32, 0, stream>>>(zb, Dw1, Db1, h1, N, DLAT, DLAT);
    dim3 g2((N_FEAT_IN + 15) / 16, (N + 15) / 16);
    wmma_gemm16<1><<<g2, 32, 0, stream>>>(h1, Dw2, Db2, recon, N, DLAT, N_FEAT_IN);
  }

  // A_pred = sigmoid(z @ z^T)  — dominant kernel, LDS-blocked fp32 WMMA
  {
    dim3 gz((N + 63) / 64, (N + 127) / 128);
    zzt_sigmoid_kernel<<<gz, 256, 0, stream>>>(zb, A_pred, N);
  }
}